// LimitedAttentionLayer_11055245820053
// MI455X (gfx1250) — compile-verified
//
#include <hip/hip_runtime.h>
#include <stdint.h>

// Problem constants (fixed by the reference).
#define BATCH    512
#define IN_FLAT  1024
#define NEURONS  4096
#define FOCUS    64

// GEMM tiling.
#define BM 32
#define BN 64
#define BK 32
#define APAD 4   // As row stride = 36 floats -> conflict-free fragment reads

typedef __attribute__((ext_vector_type(2))) float v2f;
typedef __attribute__((ext_vector_type(8))) float v8f;

// ---------------------------------------------------------------------------
// CDNA5 async global->LDS copy (ASYNCcnt-tracked). One 16-byte copy per lane.
// lds: 32-bit LDS byte offset (low 32 bits of a generic pointer to __shared__).
// g:   64-bit global address.
// ---------------------------------------------------------------------------
__device__ __forceinline__ void async_copy_b128(uint32_t lds, const float* g) {
    asm volatile("global_load_async_to_lds_b128 %0, %1, off"
                 :: "v"(lds), "v"(g) : "memory");
}

__device__ __forceinline__ void wait_asynccnt0() {
    asm volatile("s_wait_asynccnt 0x0" ::: "memory");
}

// ---------------------------------------------------------------------------
// Kernel 1: expand sparse (idx, w) into dense Wd[IN_FLAT][NEURONS] (row-major).
// One thread per neuron owns column n: zero it (coalesced across threads),
// then scatter-accumulate its 64 weights (duplicate indices handled exactly).
// ---------------------------------------------------------------------------
__global__ __launch_bounds__(256)
void expand_weights(const int* __restrict__ idx,
                    const float* __restrict__ w,
                    float* __restrict__ Wd) {
    const int n = blockIdx.x * blockDim.x + threadIdx.x;
    if (n >= NEURONS) return;

    for (int j = 0; j < IN_FLAT; ++j)
        Wd[(size_t)j * NEURONS + n] = 0.0f;

    #pragma unroll 4
    for (int f = 0; f < FOCUS; ++f) {
        const int j = idx[n * FOCUS + f];
        Wd[(size_t)j * NEURONS + n] += w[n * FOCUS + f];
    }
}

// ---------------------------------------------------------------------------
// Kernel 2: Y[512,4096] = X[512,1024] * Wd[1024,4096] + bias, exact f32 via
// V_WMMA_F32_16X16X4_F32. 256 threads = 8 wave32s (2x4 wave grid), 32x64
// block tile, K staged through double-buffered LDS with async copies.
// ---------------------------------------------------------------------------
__global__ __launch_bounds__(256)
void gemm_wmma(const float* __restrict__ X,
               const float* __restrict__ Wd,
               const float* __restrict__ bias,
               float* __restrict__ out) {
    __shared__ float As[2][BM][BK + APAD];  // 2 x 32 x 36
    __shared__ float Bs[2][BK][BN];         // 2 x 32 x 64

    const int tid  = threadIdx.x;
    const int n0   = blockIdx.x * BN;
    const int m0   = blockIdx.y * BM;

    const int wave = tid >> 5;
    const int lane = tid & 31;
    const int half = lane >> 4;   // 0: lanes 0-15, 1: lanes 16-31
    const int l16  = lane & 15;
    const int wm   = wave >> 2;   // 0..1  -> M sub-tile
    const int wn   = wave & 3;    // 0..3  -> N sub-tile

    // Tile-copy assignments (16 B per thread per buffer slot).
    const int a_row = tid >> 3;          // 0..31
    const int a_col = (tid & 7) << 2;    // 0,4,...,28
    const int b_row = tid >> 4;          // 0..15 (and +16)
    const int b_col = (tid & 15) << 2;   // 0,4,...,60

    // Per-thread LDS destinations for both buffers (generic ptr low 32 bits
    // == LDS byte offset on AMDGPU).
    uint32_t ldsA[2], ldsB0[2], ldsB1[2];
    #pragma unroll
    for (int p = 0; p < 2; ++p) {
        ldsA[p]  = (uint32_t)(uintptr_t)&As[p][a_row][a_col];
        ldsB0[p] = (uint32_t)(uintptr_t)&Bs[p][b_row][b_col];
        ldsB1[p] = (uint32_t)(uintptr_t)&Bs[p][b_row + 16][b_col];
    }

    // Per-thread global source pointers for the next chunk to issue.
    const float* gA  = X  + (size_t)(m0 + a_row) * IN_FLAT + a_col;
    const float* gB0 = Wd + (size_t)b_row        * NEURONS + n0 + b_col;
    const float* gB1 = gB0 + (size_t)16 * NEURONS;

    v8f c = {};  // f32 16x16 accumulator (8 VGPRs)

    // Prologue: stage chunk 0 into buffer 0.
    async_copy_b128(ldsA[0], gA);
    async_copy_b128(ldsB0[0], gB0);
    async_copy_b128(ldsB1[0], gB1);
    gA  += BK;
    gB0 += (size_t)BK * NEURONS;
    gB1 += (size_t)BK * NEURONS;
    wait_asynccnt0();
    __syncthreads();

    int cur = 0;
    for (int k0 = 0; k0 < IN_FLAT; k0 += BK) {
        const int nxt = cur ^ 1;

        // Issue async copies for the next chunk while we compute this one.
        if (k0 + BK < IN_FLAT) {
            async_copy_b128(ldsA[nxt], gA);
            async_copy_b128(ldsB0[nxt], gB0);
            async_copy_b128(ldsB1[nxt], gB1);
            gA  += BK;
            gB0 += (size_t)BK * NEURONS;
            gB1 += (size_t)BK * NEURONS;
        }

        // 8 WMMAs over the current K-chunk.
        // A 16x4 f32 layout: lanes 0-15 M=0..15 {K=0(v0),K=1(v1)},
        //                    lanes 16-31 M=0..15 {K=2(v0),K=3(v1)}.
        // B 4x16 f32 layout: v0 = row K=0 (lanes 0-15) / K=2 (lanes 16-31),
        //                    v1 = row K=1 / K=3.
        const float (*Asc)[BK + APAD] = As[cur];
        const float (*Bsc)[BN]        = Bs[cur];
        #pragma unroll
        for (int kk = 0; kk < BK; kk += 4) {
            const int ka = kk + 2 * half;
            v2f a, b;
            a.x = Asc[wm * 16 + l16][ka];
            a.y = Asc[wm * 16 + l16][ka + 1];
            b.x = Bsc[ka][wn * 16 + l16];
            b.y = Bsc[ka + 1][wn * 16 + l16];
            c = __builtin_amdgcn_wmma_f32_16x16x4_f32(
                /*neg_a=*/false, a, /*neg_b=*/false, b,
                /*c_mod=*/(short)0, c, /*reuse_a=*/false, /*reuse_b=*/false);
        }

        // Next chunk landed; all waves done reading `cur` before it is reused.
        wait_asynccnt0();
        __syncthreads();
        cur = nxt;
    }

    // C/D layout: VGPR v holds row (v + 8*half) for this lane's column l16.
    const int col = n0 + wn * 16 + l16;
    const float bv = bias[col];
    #pragma unroll
    for (int v = 0; v < 8; ++v) {
        const int row = m0 + wm * 16 + half * 8 + v;
        out[(size_t)row * NEURONS + col] = c[v] + bv;
    }
}

// ---------------------------------------------------------------------------
// Fallback (only if workspace is too small for the 16 MB dense matrix):
// direct gather dot-product, one thread per (batch, neuron).
// ---------------------------------------------------------------------------
__global__ __launch_bounds__(256)
void direct_gather(const float* __restrict__ x,
                   const int* __restrict__ idx,
                   const float* __restrict__ w,
                   const float* __restrict__ bias,
                   float* __restrict__ out) {
    const int g = blockIdx.x * blockDim.x + threadIdx.x;
    if (g >= BATCH * NEURONS) return;
    const int b = g / NEURONS;
    const int n = g % NEURONS;
    float acc = bias[n];
    #pragma unroll 8
    for (int f = 0; f < FOCUS; ++f)
        acc += x[(size_t)b * IN_FLAT + idx[n * FOCUS + f]] * w[n * FOCUS + f];
    out[g] = acc;
}

extern "C" void kernel_launch(void* const* d_in, const int* in_sizes, int n_in,
                              void* d_out, int out_size, void* d_ws, size_t ws_size,
                              hipStream_t stream) {
    const float* x    = (const float*)d_in[0];
    const int*   idx  = (const int*)d_in[1];
    const float* w    = (const float*)d_in[2];
    const float* bias = (const float*)d_in[3];
    float*       out  = (float*)d_out;

    const size_t need = (size_t)IN_FLAT * NEURONS * sizeof(float);  // 16 MB
    if (ws_size >= need) {
        float* Wd = (float*)d_ws;
        expand_weights<<<NEURONS / 256, 256, 0, stream>>>(idx, w, Wd);
        dim3 grid(NEURONS / BN, BATCH / BM);
        gemm_wmma<<<grid, 256, 0, stream>>>(x, Wd, bias, out);
    } else {
        direct_gather<<<(BATCH * NEURONS + 255) / 256, 256, 0, stream>>>(
            x, idx, w, bias, out);
    }
}